// GCNConvNetForRegression_27762668601834
// MI455X (gfx1250) — compile-verified
//
#include <hip/hip_runtime.h>

#define NN  40000
#define EE  640000
#define DIM 128

typedef __attribute__((ext_vector_type(2))) float v2f;
typedef __attribute__((ext_vector_type(8))) float v8f;

__device__ __forceinline__ void atomic_add_f32(float* p, float v) {
  // hardware global_atomic_add_f32 (round-to-nearest-even), no CAS loop
  unsafeAtomicAdd(p, v);
}

// ---------------- degree / dinv ----------------
__global__ __launch_bounds__(256) void deg_init(float* __restrict__ deg) {
  int i = blockIdx.x * 256 + threadIdx.x;
  if (i < NN) deg[i] = 1.0f;                       // self-loop contributes 1
}

__global__ __launch_bounds__(256) void deg_count(const int* __restrict__ dst,
                                                 float* __restrict__ deg) {
  int e = blockIdx.x * 256 + threadIdx.x;
  if (e < EE) atomic_add_f32(&deg[dst[e]], 1.0f);
}

__global__ __launch_bounds__(256) void deg_rsqrt(float* __restrict__ deg) {
  int i = blockIdx.x * 256 + threadIdx.x;
  if (i < NN) deg[i] = rsqrtf(deg[i]);             // deg -> dinv in place
}

// ---------------- fp32 WMMA GEMM: C[nrows x 128] = A[nrows x 128] @ W[128 x 128] ----------------
// block = 256 threads = 8 waves. Wave w owns output cols [16w,16w+16); block owns 64 rows
// (4 M-tiles per wave, reusing the B fragment across 4 v_wmma per K-step).
__global__ __launch_bounds__(256) void gemm128(const float* __restrict__ A,
                                               const float* __restrict__ W,
                                               float* __restrict__ C,
                                               int nrows) {
  const int lane = threadIdx.x & 31;
  const int wv   = threadIdx.x >> 5;               // 0..7 -> N tile
  const int half = lane >> 4;                      // 0/1: K sub-pair select
  const int l16  = lane & 15;
  const int row0 = blockIdx.x * 64;
  const int n0   = wv * 16;
  if (row0 >= nrows) return;                       // uniform per block: EXEC stays full

  v8f c0 = {}, c1 = {}, c2 = {}, c3 = {};

  // A frag (16x4 f32): lane = (K/2)*16 + M, vgprs hold K,K+1 -> contiguous float2
  const float* ap = A + (size_t)(row0 + l16) * DIM + 2 * half;
  // B frag (4x16 f32): lane = (K/2)*16 + N, vgpr = K%2 -> two dword loads, stride DIM
  const float* wp = W + (size_t)(2 * half) * DIM + n0 + l16;

  for (int k = 0; k < DIM; k += 4) {
    v2f b;
    b.x = wp[0];
    b.y = wp[DIM];
    v2f a0 = *(const v2f*)(ap);
    v2f a1 = *(const v2f*)(ap + 16 * DIM);
    v2f a2 = *(const v2f*)(ap + 32 * DIM);
    v2f a3 = *(const v2f*)(ap + 48 * DIM);
    c0 = __builtin_amdgcn_wmma_f32_16x16x4_f32(false, a0, false, b, (short)0, c0, false, false);
    c1 = __builtin_amdgcn_wmma_f32_16x16x4_f32(false, a1, false, b, (short)0, c1, false, false);
    c2 = __builtin_amdgcn_wmma_f32_16x16x4_f32(false, a2, false, b, (short)0, c2, false, false);
    c3 = __builtin_amdgcn_wmma_f32_16x16x4_f32(false, a3, false, b, (short)0, c3, false, false);
    ap += 4;
    wp += 4 * DIM;
  }

  // D layout: lane half selects M/M+8, VGPR r = row within half
  float* cp = C + (size_t)(row0 + half * 8) * DIM + n0 + l16;
#pragma unroll
  for (int r = 0; r < 8; ++r) {
    cp[(r)      * DIM] = c0[r];
    cp[(16 + r) * DIM] = c1[r];
    cp[(32 + r) * DIM] = c2[r];
    cp[(48 + r) * DIM] = c3[r];
  }
}

// ---------------- edge scatter: agg[dst] += h[src] * dinv[src]*dinv[dst] ----------------
// one wave per edge; lane handles 4 consecutive features (float4 gather, 4 f32 atomics)
__global__ __launch_bounds__(256) void scatter_edges(const float* __restrict__ h,
                                                     const int* __restrict__ src,
                                                     const int* __restrict__ dst,
                                                     const float* __restrict__ dinv,
                                                     float* __restrict__ agg) {
  const unsigned gid = blockIdx.x * 256u + threadIdx.x;
  const int e = __builtin_amdgcn_readfirstlane((int)(gid >> 5));  // wave-uniform -> s_loads
  if (e >= EE) return;
  const int lane = threadIdx.x & 31;
  const int s = src[e];
  const int d = dst[e];
  const float norm = dinv[s] * dinv[d];
  const float4 v = *(const float4*)(h + (size_t)s * DIM + lane * 4);
  float* out = agg + (size_t)d * DIM + lane * 4;
  atomic_add_f32(out + 0, v.x * norm);
  atomic_add_f32(out + 1, v.y * norm);
  atomic_add_f32(out + 2, v.z * norm);
  atomic_add_f32(out + 3, v.w * norm);
}

// ---------------- self-loop + bias + relu ----------------
__global__ __launch_bounds__(256) void combine_relu(const float* __restrict__ agg,
                                                    const float* __restrict__ h,
                                                    const float* __restrict__ dinv,
                                                    const float* __restrict__ bias,
                                                    float* __restrict__ act) {
  int idx = blockIdx.x * 256 + threadIdx.x;
  if (idx >= NN * DIM) return;
  int row = idx >> 7;
  int col = idx & 127;
  float di = dinv[row];
  float v = agg[idx] + h[idx] * di * di + bias[col];
  act[idx] = v > 0.0f ? v : 0.0f;
}

// ---------------- final linear: out[n] = h[n,:] . lw + lb ----------------
__global__ __launch_bounds__(256) void final_linear(const float* __restrict__ h,
                                                    const float* __restrict__ lw,
                                                    const float* __restrict__ lb,
                                                    float* __restrict__ out) {
  const unsigned gid = blockIdx.x * 256u + threadIdx.x;
  const int n = (int)(gid >> 5);
  if (n >= NN) return;
  const int lane = threadIdx.x & 31;
  const float4 hv = *(const float4*)(h + (size_t)n * DIM + lane * 4);
  const float4 wv = *(const float4*)(lw + lane * 4);
  float s = hv.x * wv.x + hv.y * wv.y + hv.z * wv.z + hv.w * wv.w;
#pragma unroll
  for (int off = 16; off > 0; off >>= 1) s += __shfl_xor(s, off, 32);
  if (lane == 0) out[n] = s + lb[0];
}

extern "C" void kernel_launch(void* const* d_in, const int* in_sizes, int n_in,
                              void* d_out, int out_size, void* d_ws, size_t ws_size,
                              hipStream_t stream) {
  const float* x  = (const float*)d_in[0];
  const int*   ei = (const int*)d_in[1];
  const float* Wl[3] = {(const float*)d_in[2], (const float*)d_in[4], (const float*)d_in[6]};
  const float* bl[3] = {(const float*)d_in[3], (const float*)d_in[5], (const float*)d_in[7]};
  const float* lw = (const float*)d_in[8];
  const float* lb = (const float*)d_in[9];
  float* out = (float*)d_out;

  const int* src = ei;        // edge_index[0]
  const int* dst = ei + EE;   // edge_index[1]

  // workspace: dinv (padded to 256KB) | act | hbuf | agg   (~62 MB total)
  float* ws   = (float*)d_ws;
  float* dinv = ws;
  float* act  = ws + 65536;
  float* hbuf = act + (size_t)NN * DIM;
  float* agg  = hbuf + (size_t)NN * DIM;

  deg_init <<<(NN + 255) / 256, 256, 0, stream>>>(dinv);
  deg_count<<<(EE + 255) / 256, 256, 0, stream>>>(dst, dinv);
  deg_rsqrt<<<(NN + 255) / 256, 256, 0, stream>>>(dinv);

  const float* hin = x;
  for (int l = 0; l < 3; ++l) {
    gemm128<<<NN / 64, 256, 0, stream>>>(hin, Wl[l], hbuf, NN);
    hipMemsetAsync(agg, 0, (size_t)NN * DIM * sizeof(float), stream);
    scatter_edges<<<(EE * 32) / 256, 256, 0, stream>>>(hbuf, src, dst, dinv, agg);
    combine_relu<<<(NN * DIM) / 256, 256, 0, stream>>>(agg, hbuf, dinv, bl[l], act);
    hin = act;
  }
  final_linear<<<(NN * 32) / 256, 256, 0, stream>>>(act, lw, lb, out);
}